// InstanceSegmentationLoss_55843164782815
// MI455X (gfx1250) — compile-verified
//
#include <hip/hip_runtime.h>
#include <hip/hip_bf16.h>

typedef __attribute__((ext_vector_type(2)))  float    v2f;
typedef __attribute__((ext_vector_type(8)))  float    v8f;
typedef __attribute__((ext_vector_type(4)))  unsigned ui4;

#define KCLS   64
#define NBINS  (KCLS * KCLS)   // 4096
#define HIST_TPB 1024
#define HIST_WGS 256

// ---------------------------------------------------------------------------
// Kernel 0: zero the global bin accumulator (d_ws is poisoned, and we
// accumulate into it every call, so this must run every launch).
// ---------------------------------------------------------------------------
__global__ void zero_bins_kernel(unsigned* __restrict__ bins) {
    int i = blockIdx.x * blockDim.x + threadIdx.x;
    if (i < NBINS) bins[i] = 0u;
}

// ---------------------------------------------------------------------------
// Kernel 1: bandwidth-bound confusion-matrix histogram.
// 16 KB LDS histogram per WG, ds_add_u32 atomics, b128 non-temporal loads,
// sparse merge into global bins.
// ---------------------------------------------------------------------------
__global__ __launch_bounds__(HIST_TPB)
void hist_kernel(const unsigned* __restrict__ pred,
                 const unsigned* __restrict__ true_m,
                 unsigned* __restrict__ bins, int n4) {
    __shared__ unsigned sbin[NBINS];
    for (int i = threadIdx.x; i < NBINS; i += HIST_TPB) sbin[i] = 0u;
    __syncthreads();

    const ui4* p4 = (const ui4*)pred;
    const ui4* t4 = (const ui4*)true_m;
    int stride = gridDim.x * HIST_TPB;
    for (int i = blockIdx.x * HIST_TPB + threadIdx.x; i < n4; i += stride) {
        ui4 p = __builtin_nontemporal_load(&p4[i]);   // global_load_b128, TH=NT
        ui4 t = __builtin_nontemporal_load(&t4[i]);
        atomicAdd(&sbin[(p.x << 6) + t.x], 1u);       // ds_add_u32
        atomicAdd(&sbin[(p.y << 6) + t.y], 1u);
        atomicAdd(&sbin[(p.z << 6) + t.z], 1u);
        atomicAdd(&sbin[(p.w << 6) + t.w], 1u);
    }
    __syncthreads();

    for (int i = threadIdx.x; i < NBINS; i += HIST_TPB) {
        unsigned v = sbin[i];
        if (v) atomicAdd(&bins[i], v);                // global_atomic_add_u32
    }
}

// ---------------------------------------------------------------------------
// Kernel 2: epilogue (single WG, 256 threads = 8 waves).
// Waves 0-3: cp = rowsum(inter_full) via chained V_WMMA_F32_16X16X4_F32
//            (A = inter tile in documented 16x4 f32 layout, B = ones).
// Waves 4-7: ct = rowsum(inter_full^T), same path.
// Then 126 threads compute IoU row/col maxes; block-reduce to the scalar loss.
// ---------------------------------------------------------------------------
__global__ __launch_bounds__(256)
void finish_kernel(const unsigned* __restrict__ bins, float* __restrict__ out) {
    __shared__ float sh_inter[NBINS];
    __shared__ float sh_cp[KCLS];
    __shared__ float sh_ct[KCLS];
    __shared__ float red_loss[256];
    __shared__ float red_cnt[256];

    const int tid  = threadIdx.x;
    const int wave = tid >> 5;
    const int lane = tid & 31;

    for (int i = tid; i < NBINS; i += 256)
        sh_inter[i] = (float)bins[i];
    __syncthreads();

    // ---- WMMA marginal sums: D = A * ones + C, chained over 16 K-tiles ----
    {
        const int  mt         = wave & 3;        // row tile (16 rows each)
        const bool transposed = (wave >= 4);     // waves 4-7 sum columns
        const int  m          = mt * 16 + (lane & 15);
        const int  khalf      = (lane >> 4) << 1;  // lanes 16-31 hold K=+2,+3

        v2f b; b.x = 1.0f; b.y = 1.0f;           // all-ones B: layout-free
        v8f c = {};
        for (int kt = 0; kt < 16; ++kt) {
            const int k0 = kt * 4 + khalf;
            v2f a;
            if (!transposed) {
                a.x = sh_inter[m * KCLS + k0];
                a.y = sh_inter[m * KCLS + k0 + 1];
            } else {
                a.x = sh_inter[k0 * KCLS + m];
                a.y = sh_inter[(k0 + 1) * KCLS + m];
            }
            // (neg_a, A, neg_b, B, c_mod, C, reuse_a, reuse_b)
            c = __builtin_amdgcn_wmma_f32_16x16x4_f32(
                    false, a, false, b, (short)0, c, false, false);
        }
        // D layout: VGPR v, lanes 0-15 -> M=v (N=lane); lanes 16-31 -> M=8+v.
        // Sum is replicated across N; extract from the N==0 column.
        float* dst = transposed ? sh_ct : sh_cp;
        if (lane == 0) {
            #pragma unroll
            for (int v = 0; v < 8; ++v) dst[mt * 16 + v] = c[v];
        }
        if (lane == 16) {
            #pragma unroll
            for (int v = 0; v < 8; ++v) dst[mt * 16 + 8 + v] = c[v];
        }
    }
    __syncthreads();

    // ---- IoU maxes over the 63x63 submatrix (labels 1..63) ----
    float loss = 0.0f, cnt = 0.0f;
    if (tid < 63) {                                 // pred side: row i
        const int i = tid + 1;
        const float cpi = sh_cp[i];
        if (cpi > 0.0f) {
            float mx = 0.0f;
            for (int j = 1; j < KCLS; ++j) {
                const float in_ = sh_inter[i * KCLS + j];
                const float un  = cpi + sh_ct[j] - in_;
                const float iou = (un > 0.0f) ? in_ / fmaxf(un, 1.0f) : 0.0f;
                mx = fmaxf(mx, iou);
            }
            loss = 1.0f - mx;
            cnt  = 1.0f;
        }
    } else if (tid >= 64 && tid < 127) {            // true side: col j
        const int j = tid - 63;
        const float ctj = sh_ct[j];
        if (ctj > 0.0f) {
            float mx = 0.0f;
            for (int i = 1; i < KCLS; ++i) {
                const float in_ = sh_inter[i * KCLS + j];
                const float un  = sh_cp[i] + ctj - in_;
                const float iou = (un > 0.0f) ? in_ / fmaxf(un, 1.0f) : 0.0f;
                mx = fmaxf(mx, iou);
            }
            loss = 1.0f - mx;
            cnt  = 1.0f;
        }
    }

    red_loss[tid] = loss;
    red_cnt[tid]  = cnt;
    __syncthreads();
    for (int s = 128; s > 0; s >>= 1) {
        if (tid < s) {
            red_loss[tid] += red_loss[tid + s];
            red_cnt[tid]  += red_cnt[tid + s];
        }
        __syncthreads();
    }
    if (tid == 0) {
        const float n = red_cnt[0];
        out[0] = (n > 0.0f) ? red_loss[0] / fmaxf(n, 1.0f) : 0.0f;
    }
}

// ---------------------------------------------------------------------------
extern "C" void kernel_launch(void* const* d_in, const int* in_sizes, int n_in,
                              void* d_out, int out_size, void* d_ws, size_t ws_size,
                              hipStream_t stream) {
    const unsigned* pred   = (const unsigned*)d_in[0];
    const unsigned* true_m = (const unsigned*)d_in[1];
    unsigned* bins = (unsigned*)d_ws;      // 4096 * 4 B = 16 KB of scratch
    float*    out  = (float*)d_out;

    const int n  = in_sizes[0];            // 4096*4096 = 16,777,216
    const int n4 = n >> 2;                 // uint4 elements

    zero_bins_kernel<<<(NBINS + 1023) / 1024, 1024, 0, stream>>>(bins);
    hist_kernel<<<HIST_WGS, HIST_TPB, 0, stream>>>(pred, true_m, bins, n4);
    finish_kernel<<<1, 256, 0, stream>>>(bins, out);
}